// GCN_60945585930893
// MI455X (gfx1250) — compile-verified
//
#include <hip/hip_runtime.h>

typedef float v2f __attribute__((ext_vector_type(2)));
typedef float v4f __attribute__((ext_vector_type(4)));
typedef float v8f __attribute__((ext_vector_type(8)));

#define B_   32
#define N_   4096
#define IN_  12
#define HID_ 256
#define OUT_ 128

// ---------------------------------------------------------------------------
// Kernel 1: per-batch mean over the node axis (exploits mean/matmul
// commutation) fused with the tiny K=12 first-layer GEMM.
//   H1[b,h] = b1[h] + sum_f (mean_n x[b,n,f]) * W1[f,h]
// One block per batch. Node rows are 48 bytes -> three aligned float4 loads.
// ---------------------------------------------------------------------------
__global__ __launch_bounds__(256) void gcn_reduce_l1(
    const float* __restrict__ x, const float* __restrict__ W1,
    const float* __restrict__ b1, float* __restrict__ H1) {
  __shared__ float red[256][IN_];
  const int b = blockIdx.x;
  const int t = threadIdx.x;
  const float* base = x + (size_t)b * N_ * IN_;

  v4f a0 = {}, a1 = {}, a2 = {};
  for (int n = t; n < N_; n += 256) {
    const v4f* p = (const v4f*)(base + n * IN_);
    a0 += p[0];
    a1 += p[1];
    a2 += p[2];
  }
  red[t][0] = a0.x;  red[t][1] = a0.y;  red[t][2]  = a0.z;  red[t][3]  = a0.w;
  red[t][4] = a1.x;  red[t][5] = a1.y;  red[t][6]  = a1.z;  red[t][7]  = a1.w;
  red[t][8] = a2.x;  red[t][9] = a2.y;  red[t][10] = a2.z;  red[t][11] = a2.w;
  __syncthreads();

  for (int s = 128; s > 0; s >>= 1) {
    if (t < s) {
#pragma unroll
      for (int f = 0; f < IN_; ++f) red[t][f] += red[t + s][f];
    }
    __syncthreads();
  }

  // Fused layer-1 GEMM: one HID channel per thread, K = 12.
  float h = b1[t];
#pragma unroll
  for (int f = 0; f < IN_; ++f)
    h += (red[0][f] * (1.0f / (float)N_)) * W1[f * HID_ + t];
  H1[b * HID_ + t] = h;
}

// ---------------------------------------------------------------------------
// Kernel 2: H2[32,128] = H1[32,256] @ W2[256,128] + b2, via fp32 WMMA.
// One 512-thread block = 16 waves; wave (mi,ni) owns the 16x16 tile at
// rows mi*16.., cols ni*16.., accumulating 64 K-steps of 4 with
// V_WMMA_F32_16X16X4_F32 (exact fp32 math, matches reference precision).
//
// Fragment layouts per CDNA5 ISA 7.12.2 (wave32):
//   A 16x4 f32 : lane L -> row M=L%16; VGPR0 holds K = (L<16 ? 0 : 2)+k0,
//                VGPR1 holds K+1.
//   B 4x16 f32 : lane L -> col N=L%16; VGPR0 holds K row (L<16 ? 0 : 2)+k0,
//                VGPR1 holds K+1.
//   C/D 16x16  : lane L -> col N=L%16; VGPR v -> row M = v + (L<16 ? 0 : 8).
// ---------------------------------------------------------------------------
__global__ __launch_bounds__(512) void gcn_l2_wmma(
    const float* __restrict__ H1, const float* __restrict__ W2,
    const float* __restrict__ b2, float* __restrict__ H2) {
  const int t     = threadIdx.x;
  const int wave  = t >> 5;
  const int lane  = t & 31;
  const int mi    = wave >> 3;     // 0..1  (M tiles of 32 rows)
  const int ni    = wave & 7;      // 0..7  (N tiles of 128 cols)
  const int laneM = lane & 15;
  const int hi    = lane >> 4;     // lane half selects K pair / M half
  const int kOff  = 2 * hi;

  const float* arow = H1 + (size_t)(mi * 16 + laneM) * HID_;
  const float* bcol = W2 + ni * 16 + laneM;

  v8f c = {};
  for (int k0 = 0; k0 < HID_; k0 += 4) {
    v2f a, bm;
    a.x  = arow[k0 + kOff];
    a.y  = arow[k0 + kOff + 1];
    bm.x = bcol[(size_t)(k0 + kOff) * OUT_];
    bm.y = bcol[(size_t)(k0 + kOff + 1) * OUT_];
    // 8 args: (neg_a, A, neg_b, B, c_mod, C, reuse_a, reuse_b)
    c = __builtin_amdgcn_wmma_f32_16x16x4_f32(
        false, a, false, bm, (short)0, c, false, false);
  }

  const int   col  = ni * 16 + laneM;
  const float bias = b2[col];
#pragma unroll
  for (int v = 0; v < 8; ++v) {
    const int row = mi * 16 + v + 8 * hi;
    H2[row * OUT_ + col] = c[v] + bias;
  }
}

// ---------------------------------------------------------------------------
// Kernel 3: broadcast H2[32,128] across the 4096-node axis -> out[32,4096,128].
// 64 MB of pure streaming stores: 128-bit non-temporal stores so the
// write-once output doesn't pollute WGP$/L2. 64 nodes per 256-thread block.
// ---------------------------------------------------------------------------
__global__ __launch_bounds__(256) void gcn_broadcast(
    const float* __restrict__ H2, float* __restrict__ out) {
  const int b       = blockIdx.x >> 6;   // batch
  const int chunk   = blockIdx.x & 63;   // 64-node chunk within batch
  const int t       = threadIdx.x;
  const int col4    = (t & 31) << 2;     // 32 lanes cover 128 cols in float4s
  const int nodeOff = t >> 5;            // 8 node-rows per iteration

  const v4f val = *(const v4f*)(H2 + b * OUT_ + col4);
  float* obase = out + ((size_t)b * N_ + (size_t)chunk * 64) * OUT_;
#pragma unroll
  for (int i = 0; i < 8; ++i) {
    v4f* dst = (v4f*)(obase + (size_t)(nodeOff + i * 8) * OUT_ + col4);
    __builtin_nontemporal_store(val, dst);
  }
}

// ---------------------------------------------------------------------------
extern "C" void kernel_launch(void* const* d_in, const int* in_sizes, int n_in,
                              void* d_out, int out_size, void* d_ws,
                              size_t ws_size, hipStream_t stream) {
  const float* x  = (const float*)d_in[0];   // [32,4096,12]
  const float* W1 = (const float*)d_in[1];   // [12,256]
  const float* b1 = (const float*)d_in[2];   // [256]
  const float* W2 = (const float*)d_in[3];   // [256,128]
  const float* b2 = (const float*)d_in[4];   // [128]
  float* out = (float*)d_out;                // [32,4096,128]

  float* H1 = (float*)d_ws;                  // [32,256]
  float* H2 = H1 + B_ * HID_;                // [32,128]

  gcn_reduce_l1<<<B_, 256, 0, stream>>>(x, W1, b1, H1);
  gcn_l2_wmma<<<1, 512, 0, stream>>>(H1, W2, b2, H2);
  gcn_broadcast<<<B_ * 64, 256, 0, stream>>>(H2, out);
}